// EpsilonState_24318104830347
// MI455X (gfx1250) — compile-verified
//
#include <hip/hip_runtime.h>
#include <math.h>

// ---------------------------------------------------------------------------
// Problem dims (from reference): n=32 spins, N=2n=64, Pfaffian dim 2N=128.
// ---------------------------------------------------------------------------
#define NS        32
#define NDIM      64
#define PDIM      128
#define NBATCH    64
#define NTHREADS  256
#define NWAVES    (NTHREADS / 32)

typedef __attribute__((ext_vector_type(2))) float v2f;
typedef __attribute__((ext_vector_type(8))) float v8f;

#define WMMA4(a, b, c)                                                         \
  __builtin_amdgcn_wmma_f32_16x16x4_f32(false, (a), false, (b), (short)0, (c), \
                                        false, false)

__device__ inline v8f v8zero() { v8f z = {0.f,0.f,0.f,0.f,0.f,0.f,0.f,0.f}; return z; }

// ------------------------- complex double helpers --------------------------
struct dc { double re, im; };
__device__ inline dc cmul(dc a, dc b) { return dc{a.re*b.re - a.im*b.im, a.re*b.im + a.im*b.re}; }
__device__ inline dc cadd(dc a, dc b) { return dc{a.re + b.re, a.im + b.im}; }
__device__ inline dc csub(dc a, dc b) { return dc{a.re - b.re, a.im - b.im}; }
__device__ inline dc conj_(dc a)      { return dc{a.re, -a.im}; }
__device__ inline dc cscale(dc a, double s) { return dc{a.re*s, a.im*s}; }
__device__ inline double cabs2(dc a)  { return a.re*a.re + a.im*a.im; }
__device__ inline dc cdiv(dc a, dc b) {
  double d = cabs2(b);
  return dc{(a.re*b.re + a.im*b.im)/d, (a.im*b.re - a.re*b.im)/d};
}
__device__ inline dc clog_(dc a) { return dc{0.5*log(cabs2(a)), atan2(a.im, a.re)}; }
__device__ inline dc cexp_(dc a) { double e = exp(a.re); return dc{e*cos(a.im), e*sin(a.im)}; }

// ------------------------- workspace layout (doubles) ----------------------
// Front of d_ws: published precompute results.
//   [0      .. 8191 ]  Ghz_plus   (64x64 dc)
//   [8192   .. 16383]  Ghz_minus  (64x64 dc)
//   [16384  .. 16387]  logeta_Ghz_plus, logeta_Ghz_minus (2 x dc)
#define SHARED_DBL 16640
// Per-block arena (doubles).  The 128x128 Pfaffian matrix now lives in LDS,
// so the arena only holds GEMM operands + real series temps + f32 planes.
#define RT_OFF    0        // 4 real 64x64 temps (A,P1,P2,P3) = 16384
#define GS_OFF    16384    // G / gbar / Gz 64x64 dc = 8192
#define ST_OFF    24576    // state / R  64x32 dc = 4096
#define VX_OFF    28672    // v          64x32 dc = 4096
#define LH_OFF    32768    // Lh         32x64 dc = 4096
#define MA_OFF    36864    // Maug       32x96 dc = 6144
#define X_OFF     43008    // X          32x64 dc = 4096
#define RES_OFF   47104    // pfaffian sign (dc) + logabs (1) -> 4
#define E1_OFF    47360    // expH real 64x64 = 4096
#define E2_OFF    51456    // Gh   real 64x64 = 4096
#define FS_OFF    55552    // f32 hi/lo planes: 16384 floats = 8192 doubles
#define ARENA_DBL 65536

// ===========================================================================
// WMMA GEMMs through V_WMMA_F32_16X16X4_F32 with double-float (hi/lo) split.
// Phase 1: cooperatively split operands into f32 hi/lo planes in the per-block
//          float scratch (B stored transposed, K-major, so both fragments are
//          contiguous 64-bit loads).
// Phase 2: per-wave 16x16 tiles, pure v2f loads + WMMA chains; f64 recombine.
// All M,N multiples of 16; K multiple of 4; EXEC all-ones per active wave.
// ===========================================================================

// real C(MxN,ldc) = A(MxK) * B(KxN)
__device__ __forceinline__ void mm_real(const double* __restrict__ A,
                                        const double* __restrict__ B,
                                        double* __restrict__ C,
                                        int M, int K, int N, int ldc,
                                        float* __restrict__ fs)
{
  __syncthreads();
  const int tid = threadIdx.x;
  const int MK = M * K, KN = K * N;
  float *Ah = fs, *Al = Ah + MK, *Bh = Al + MK, *Bl = Bh + KN;
  for (int e = tid; e < MK; e += NTHREADS) {
    double a = A[e];
    float h = (float)a; Ah[e] = h; Al[e] = (float)(a - (double)h);
  }
  for (int e = tid; e < KN; e += NTHREADS) {
    const int k = e / N, c = e % N;          // transpose: [c*K + k]
    double b = B[e];
    float h = (float)b; Bh[c*K + k] = h; Bl[c*K + k] = (float)(b - (double)h);
  }
  __syncthreads();
  const int wid = tid >> 5, lane = tid & 31;
  const int l15 = lane & 15;
  const int koff   = (lane < 16) ? 0 : 2;
  const int rowoff = (lane < 16) ? 0 : 8;
  const int tilesM = M >> 4, tilesN = N >> 4;
  for (int t = wid; t < tilesM * tilesN; t += NWAVES) {
    const int tm = (t / tilesN) << 4;
    const int tn = (t % tilesN) << 4;
    const int abase = (tm + l15) * K + koff;
    const int bbase = (tn + l15) * K + koff;
    v8f hh = v8zero(), cc = v8zero();
    for (int kc = 0; kc < K; kc += 4) {
      v2f ah = *(const v2f*)(Ah + abase + kc), al = *(const v2f*)(Al + abase + kc);
      v2f bh = *(const v2f*)(Bh + bbase + kc), bl = *(const v2f*)(Bl + bbase + kc);
      hh = WMMA4(ah, bh, hh);
      cc = WMMA4(ah, bl, cc);
      cc = WMMA4(al, bh, cc);
    }
#pragma unroll
    for (int r = 0; r < 8; ++r)
      C[(tm + rowoff + r) * ldc + (tn + l15)] = (double)hh[r] + (double)cc[r];
  }
  __syncthreads();
}

// complex C(MxN,ldc) = A(MxK) * B(KxN), all complex
__device__ __forceinline__ void mm_cplx(const dc* __restrict__ A,
                                        const dc* __restrict__ B,
                                        dc* __restrict__ C,
                                        int M, int K, int N, int ldc,
                                        float* __restrict__ fs)
{
  __syncthreads();
  const int tid = threadIdx.x;
  const int MK = M * K, KN = K * N;
  float *Arh = fs,       *Arl = Arh + MK, *Aih = Arl + MK, *Ail = Aih + MK;
  float *Brh = Ail + MK, *Brl = Brh + KN, *Bih = Brl + KN, *Bil = Bih + KN;
  for (int e = tid; e < MK; e += NTHREADS) {
    dc a = A[e];
    float h = (float)a.re; Arh[e] = h; Arl[e] = (float)(a.re - (double)h);
    h = (float)a.im;       Aih[e] = h; Ail[e] = (float)(a.im - (double)h);
  }
  for (int e = tid; e < KN; e += NTHREADS) {
    const int k = e / N, c = e % N;          // transpose: [c*K + k]
    dc b = B[e];
    float h = (float)b.re; Brh[c*K+k] = h; Brl[c*K+k] = (float)(b.re - (double)h);
    h = (float)b.im;       Bih[c*K+k] = h; Bil[c*K+k] = (float)(b.im - (double)h);
  }
  __syncthreads();
  const int wid = tid >> 5, lane = tid & 31;
  const int l15 = lane & 15;
  const int koff   = (lane < 16) ? 0 : 2;
  const int rowoff = (lane < 16) ? 0 : 8;
  const int tilesM = M >> 4, tilesN = N >> 4;
  for (int t = wid; t < tilesM * tilesN; t += NWAVES) {
    const int tm = (t / tilesN) << 4;
    const int tn = (t % tilesN) << 4;
    const int abase = (tm + l15) * K + koff;
    const int bbase = (tn + l15) * K + koff;
    v8f rr_h = v8zero(), rr_c = v8zero(), ii_h = v8zero(), ii_c = v8zero();
    v8f ri_h = v8zero(), ri_c = v8zero(), ir_h = v8zero(), ir_c = v8zero();
    for (int kc = 0; kc < K; kc += 4) {
      v2f arh = *(const v2f*)(Arh + abase + kc), arl = *(const v2f*)(Arl + abase + kc);
      v2f aih = *(const v2f*)(Aih + abase + kc), ail = *(const v2f*)(Ail + abase + kc);
      v2f brh = *(const v2f*)(Brh + bbase + kc), brl = *(const v2f*)(Brl + bbase + kc);
      v2f bih = *(const v2f*)(Bih + bbase + kc), bil = *(const v2f*)(Bil + bbase + kc);
      rr_h = WMMA4(arh, brh, rr_h);
      rr_c = WMMA4(arh, brl, rr_c); rr_c = WMMA4(arl, brh, rr_c);
      ii_h = WMMA4(aih, bih, ii_h);
      ii_c = WMMA4(aih, bil, ii_c); ii_c = WMMA4(ail, bih, ii_c);
      ri_h = WMMA4(arh, bih, ri_h);
      ri_c = WMMA4(arh, bil, ri_c); ri_c = WMMA4(arl, bih, ri_c);
      ir_h = WMMA4(aih, brh, ir_h);
      ir_c = WMMA4(aih, brl, ir_c); ir_c = WMMA4(ail, brh, ir_c);
    }
#pragma unroll
    for (int r = 0; r < 8; ++r) {
      const int row = tm + rowoff + r, col = tn + l15;
      double re = ((double)rr_h[r] + (double)rr_c[r]) - ((double)ii_h[r] + (double)ii_c[r]);
      double im = ((double)ri_h[r] + (double)ri_c[r]) + ((double)ir_h[r] + (double)ir_c[r]);
      C[row * ldc + col] = dc{re, im};
    }
  }
  __syncthreads();
}

// complex C(MxN,ldc) = Areal(MxK) * B(KxN complex)
__device__ __forceinline__ void mm_rc(const double* __restrict__ A,
                                      const dc* __restrict__ B,
                                      dc* __restrict__ C,
                                      int M, int K, int N, int ldc,
                                      float* __restrict__ fs)
{
  __syncthreads();
  const int tid = threadIdx.x;
  const int MK = M * K, KN = K * N;
  float *Ah = fs, *Al = Ah + MK;
  float *Brh = Al + MK, *Brl = Brh + KN, *Bih = Brl + KN, *Bil = Bih + KN;
  for (int e = tid; e < MK; e += NTHREADS) {
    double a = A[e];
    float h = (float)a; Ah[e] = h; Al[e] = (float)(a - (double)h);
  }
  for (int e = tid; e < KN; e += NTHREADS) {
    const int k = e / N, c = e % N;
    dc b = B[e];
    float h = (float)b.re; Brh[c*K+k] = h; Brl[c*K+k] = (float)(b.re - (double)h);
    h = (float)b.im;       Bih[c*K+k] = h; Bil[c*K+k] = (float)(b.im - (double)h);
  }
  __syncthreads();
  const int wid = tid >> 5, lane = tid & 31;
  const int l15 = lane & 15;
  const int koff   = (lane < 16) ? 0 : 2;
  const int rowoff = (lane < 16) ? 0 : 8;
  const int tilesM = M >> 4, tilesN = N >> 4;
  for (int t = wid; t < tilesM * tilesN; t += NWAVES) {
    const int tm = (t / tilesN) << 4;
    const int tn = (t % tilesN) << 4;
    const int abase = (tm + l15) * K + koff;
    const int bbase = (tn + l15) * K + koff;
    v8f re_h = v8zero(), re_c = v8zero(), im_h = v8zero(), im_c = v8zero();
    for (int kc = 0; kc < K; kc += 4) {
      v2f ah  = *(const v2f*)(Ah  + abase + kc), al  = *(const v2f*)(Al  + abase + kc);
      v2f brh = *(const v2f*)(Brh + bbase + kc), brl = *(const v2f*)(Brl + bbase + kc);
      v2f bih = *(const v2f*)(Bih + bbase + kc), bil = *(const v2f*)(Bil + bbase + kc);
      re_h = WMMA4(ah, brh, re_h);
      re_c = WMMA4(ah, brl, re_c); re_c = WMMA4(al, brh, re_c);
      im_h = WMMA4(ah, bih, im_h);
      im_c = WMMA4(ah, bil, im_c); im_c = WMMA4(al, bih, im_c);
    }
#pragma unroll
    for (int r = 0; r < 8; ++r) {
      const int row = tm + rowoff + r, col = tn + l15;
      C[row * ldc + col] = dc{(double)re_h[r] + (double)re_c[r],
                              (double)im_h[r] + (double)im_c[r]};
    }
  }
  __syncthreads();
}

// ===========================================================================
// Gauss-Jordan with partial pivoting on Maug (32 x 96); X in columns 32..95.
// ===========================================================================
__device__ void solve32(dc* __restrict__ Ma)
{
  __shared__ int s_piv;
  __shared__ dc  s_colk[32];
  const int tid = threadIdx.x;
  for (int k = 0; k < 32; ++k) {
    __syncthreads();
    if (tid == 0) {
      int best = k; double bv = cabs2(Ma[k * 96 + k]);
      for (int r = k + 1; r < 32; ++r) {
        double v = cabs2(Ma[r * 96 + k]);
        if (v > bv) { bv = v; best = r; }
      }
      s_piv = best;
    }
    __syncthreads();
    const int p = s_piv;
    if (p != k) {
      for (int c = tid; c < 96; c += NTHREADS) {
        dc t = Ma[k * 96 + c]; Ma[k * 96 + c] = Ma[p * 96 + c]; Ma[p * 96 + c] = t;
      }
    }
    __syncthreads();
    const dc piv = Ma[k * 96 + k];
    __syncthreads();
    for (int c = tid; c < 96; c += NTHREADS)
      Ma[k * 96 + c] = cdiv(Ma[k * 96 + c], piv);
    __syncthreads();
    if (tid < 32) s_colk[tid] = Ma[tid * 96 + k];
    __syncthreads();
    for (int e = tid; e < 32 * 96; e += NTHREADS) {
      const int r = e / 96;
      if (r == k) continue;
      Ma[e] = csub(Ma[e], cmul(s_colk[r], Ma[k * 96 + (e % 96)]));
    }
    __syncthreads();
  }
}

// ===========================================================================
// G = I - 2 * R @ solve(L^H @ R, L^H)     (L,R: 64x32 ; G: 64x64)
// ===========================================================================
__device__ void green_from_two(const dc* __restrict__ L, const dc* __restrict__ R,
                               dc* __restrict__ G, dc* __restrict__ Lh,
                               dc* __restrict__ Ma, dc* __restrict__ X,
                               float* __restrict__ fs)
{
  const int tid = threadIdx.x;
  for (int e = tid; e < NS * NDIM; e += NTHREADS) {
    const int i = e / NDIM, k = e % NDIM;
    Lh[e] = conj_(L[k * NS + i]);
  }
  mm_cplx(Lh, R, Ma, 32, NDIM, NS, 96, fs);        // M -> Ma[:,0:32]
  for (int e = tid; e < NS * NDIM; e += NTHREADS) {
    const int i = e / NDIM, k = e % NDIM;
    Ma[i * 96 + 32 + k] = Lh[e];                    // RHS -> Ma[:,32:96]
  }
  __syncthreads();
  solve32(Ma);
  for (int e = tid; e < NS * NDIM; e += NTHREADS) {
    const int i = e / NDIM, k = e % NDIM;
    X[e] = Ma[i * 96 + 32 + k];
  }
  mm_cplx(R, X, G, NDIM, NS, NDIM, NDIM, fs);      // gbar
  for (int e = tid; e < NDIM * NDIM; e += NTHREADS) {
    const int r = e / NDIM, c = e % NDIM;
    dc g = G[e];
    G[e] = dc{(r == c ? 1.0 : 0.0) - 2.0 * g.re, -2.0 * g.im};
  }
  __syncthreads();
}

// ===========================================================================
// pfmat = [[ (G1-G1^T)/2 , -I ], [ I , (G2-G2^T)/2 ]]  (128x128, into LDS)
// ===========================================================================
__device__ __forceinline__ void build_pfmat(const dc* __restrict__ G1,
                                            const dc* __restrict__ G2,
                                            dc* __restrict__ P)
{
  for (int e = threadIdx.x; e < PDIM * PDIM; e += NTHREADS) {
    const int r = e >> 7, c = e & 127;
    dc val;
    if (r < NDIM && c < NDIM) {
      dc a = G1[r * NDIM + c], b = G1[c * NDIM + r];
      val = dc{0.5 * (a.re - b.re), 0.5 * (a.im - b.im)};
    } else if (r < NDIM) {
      val = dc{(c - NDIM) == r ? -1.0 : 0.0, 0.0};
    } else if (c < NDIM) {
      val = dc{(r - NDIM) == c ? 1.0 : 0.0, 0.0};
    } else {
      dc a = G2[(r - NDIM) * NDIM + (c - NDIM)], b = G2[(c - NDIM) * NDIM + (r - NDIM)];
      val = dc{0.5 * (a.re - b.re), 0.5 * (a.im - b.im)};
    }
    P[e] = val;
  }
  __syncthreads();
}

// variant with real G1 (precompute path)
__device__ __forceinline__ void build_pfmat_r(const double* __restrict__ G1,
                                              const dc* __restrict__ G2,
                                              dc* __restrict__ P)
{
  for (int e = threadIdx.x; e < PDIM * PDIM; e += NTHREADS) {
    const int r = e >> 7, c = e & 127;
    dc val;
    if (r < NDIM && c < NDIM) {
      val = dc{0.5 * (G1[r * NDIM + c] - G1[c * NDIM + r]), 0.0};
    } else if (r < NDIM) {
      val = dc{(c - NDIM) == r ? -1.0 : 0.0, 0.0};
    } else if (c < NDIM) {
      val = dc{(r - NDIM) == c ? 1.0 : 0.0, 0.0};
    } else {
      dc a = G2[(r - NDIM) * NDIM + (c - NDIM)], b = G2[(c - NDIM) * NDIM + (r - NDIM)];
      val = dc{0.5 * (a.re - b.re), 0.5 * (a.im - b.im)};
    }
    P[e] = val;
  }
  __syncthreads();
}

// ===========================================================================
// slog-Pfaffian: 126-step Householder scan on the LDS-resident 128x128
// matrix.  All vectors (hv, hw, matvec partials) also in LDS; the matvec
// uses 2 threads/row so all 256 lanes work in the dominant loop.
// ===========================================================================
__device__ __forceinline__ void slog_pfaffian(dc* __restrict__ A,
                                              dc* __restrict__ sign_out,
                                              double* __restrict__ la_out)
{
  __shared__ double s_red[NTHREADS];
  __shared__ dc s_mv[NTHREADS];
  __shared__ dc s_hv[PDIM];
  __shared__ dc s_hw[PDIM];
  __shared__ dc s_sign, s_vn, s_alpha;
  __shared__ double s_logpf, s_invnorm, s_tau;
  __shared__ int s_cond;
  const int tid = threadIdx.x;
  if (tid == 0) { s_sign = dc{1.0, 0.0}; s_logpf = 0.0; }
  __syncthreads();

  for (int i = 0; i < PDIM - 2; ++i) {
    const int nh = i + 1;
    double part = 0.0;
    for (int k = tid; k < PDIM; k += NTHREADS)
      if (k > nh) part += cabs2(A[k * PDIM + i]);
    s_red[tid] = part; __syncthreads();
    for (int s = NTHREADS / 2; s > 0; s >>= 1) {
      if (tid < s) s_red[tid] += s_red[tid + s];
      __syncthreads();
    }
    if (tid == 0) {
      const double sigma = s_red[0];
      const dc xn = A[nh * PDIM + i];
      if (sigma == 0.0) {
        s_cond = 1; s_tau = 0.0; s_alpha = xn; s_vn = dc{0.0, 0.0}; s_invnorm = 0.0;
      } else {
        const double norm_x = sqrt(cabs2(xn) + sigma);
        const double axn = sqrt(cabs2(xn));
        const dc phase = (axn == 0.0) ? dc{1.0, 0.0} : cscale(xn, 1.0 / axn);
        const dc vn = cadd(xn, cscale(phase, norm_x));
        s_alpha = cscale(phase, -norm_x);
        s_invnorm = 1.0 / sqrt(cabs2(vn) + sigma);
        s_vn = vn; s_tau = 2.0; s_cond = 0;
      }
    }
    __syncthreads();
    const int cond = s_cond;
    for (int k = tid; k < PDIM; k += NTHREADS) {
      dc val = dc{0.0, 0.0};
      if (!cond) {
        if (k == nh)     val = cscale(s_vn, s_invnorm);
        else if (k > nh) val = cscale(A[k * PDIM + i], s_invnorm);
      }
      s_hv[k] = val;
    }
    __syncthreads();
    {   // hw = tau * A @ conj(hv) : 2 threads per row
      const int r = tid >> 1, half = tid & 1;
      dc acc = dc{0.0, 0.0};
      if (!cond) {
        for (int k = nh + half; k < PDIM; k += 2)
          acc = cadd(acc, cmul(A[r * PDIM + k], conj_(s_hv[k])));
      }
      s_mv[tid] = acc;
      __syncthreads();
      if (half == 0) {
        if (cond) {
          s_hw[r] = dc{0.0, 0.0};
        } else {
          s_hw[r] = cscale(cadd(s_mv[tid], s_mv[tid + 1]), s_tau);
        }
      }
      __syncthreads();
    }
    if (!cond) {
      for (int e = tid; e < PDIM * PDIM; e += NTHREADS) {
        const int r = e >> 7, c = e & 127;
        A[e] = cadd(A[e], csub(cmul(s_hv[r], s_hw[c]), cmul(s_hw[r], s_hv[c])));
      }
    }
    __syncthreads();
    if (tid == 0) {
      const double om = 1.0 - s_tau;
      double l = log(fabs(om));
      dc sgn = cscale(s_sign, om / fabs(om));
      if ((i & 1) == 0) {
        const dc alpha = s_alpha;
        const double aa = sqrt(cabs2(alpha));
        l += log(aa);
        sgn = cmul(sgn, dc{-alpha.re / aa, -alpha.im / aa});
      }
      s_logpf += l;
      s_sign = sgn;
    }
    __syncthreads();
  }
  if (tid == 0) {
    const dc last = A[(PDIM - 2) * PDIM + (PDIM - 1)];
    const double aa = sqrt(cabs2(last));
    *la_out   = s_logpf + log(aa);
    *sign_out = cmul(s_sign, cscale(last, 1.0 / aa));
  }
  __syncthreads();
}

// ===========================================================================
// build v(s, PX) per reference gen_v_from_zz_px, into 64x32 complex
// ===========================================================================
__device__ void build_v(const double* __restrict__ s, int PXv, dc* __restrict__ V)
{
  const int tid = threadIdx.x;
  for (int e = tid; e < NDIM * NS; e += NTHREADS) V[e] = dc{0.0, 0.0};
  __syncthreads();
  const double inv_s = 0.70710678118654752440;
  for (int k = tid; k < NS; k += NTHREADS) {
    const double zz = s[k] * s[(k + 1) % NS];
    double sg = (zz > 0.0) ? 1.0 : ((zz < 0.0) ? -1.0 : 0.0);
    if (k == NS - 1) sg = -(double)PXv * sg;
    const int arow = 2 * k + 1, brow = (2 * k + 2) % NDIM;
    V[arow * NS + k] = dc{0.0, -sg * inv_s};   // -1j * sgn / sqrt(2)
    V[brow * NS + k] = dc{inv_s, 0.0};
  }
  __syncthreads();
}

// ===========================================================================
// Precompute kernel: block 0 -> (H1, PX=+1), block 1 -> (H2, PX=-1).
// Series replace eigh:  expH = exp(A),  Gh = -tanh(A/2),
// logeta = 0.5 tr(B^2) - (1/12) tr(B^4),  B = A/2  (||A|| ~ 1e-3).
// ===========================================================================
__global__ void precompute_kernel(const double* __restrict__ H1,
                                  const double* __restrict__ H2,
                                  const double* __restrict__ s0,
                                  double* __restrict__ ws)
{
  __shared__ dc sP[PDIM * PDIM];             // 256 KB LDS Pfaffian matrix
  const int px = blockIdx.x;                 // 0: plus / H1, 1: minus / H2
  const int PXv = (px == 0) ? 1 : -1;
  const double* H = (px == 0) ? H1 : H2;
  double* shr = ws;
  double* ar  = ws + SHARED_DBL + (size_t)(NBATCH + px) * ARENA_DBL;

  double* A  = ar + RT_OFF;
  double* P1 = A + 4096;
  double* P2 = P1 + 4096;
  double* P3 = P2 + 4096;
  double* E  = ar + E1_OFF;                  // expH
  double* Gh = ar + E2_OFF;                  // green (real)
  dc* V  = (dc*)(ar + VX_OFF);
  dc* R  = (dc*)(ar + ST_OFF);
  dc* Lh = (dc*)(ar + LH_OFF);
  dc* Ma = (dc*)(ar + MA_OFF);
  dc* X  = (dc*)(ar + X_OFF);
  dc* Gz = (dc*)(ar + GS_OFF);
  dc* resSign = (dc*)(ar + RES_OFF);
  double* resLa = ar + RES_OFF + 2;
  float* fs = (float*)(ar + FS_OFF);

  __shared__ double s_red[NTHREADS], s_red2[NTHREADS];
  __shared__ double s_le;
  const int tid = threadIdx.x;

  // A = (H - H^T)/2
  for (int e = tid; e < NDIM * NDIM; e += NTHREADS) {
    const int r = e / NDIM, c = e % NDIM;
    A[e] = 0.5 * (H[e] - H[c * NDIM + r]);
  }
  __syncthreads();

  // expH = exp(A) by Horner, 9 terms (WMMA chain)
  for (int e = tid; e < NDIM * NDIM; e += NTHREADS)
    E[e] = ((e / NDIM) == (e % NDIM)) ? 1.0 : 0.0;
  for (int m = 9; m >= 1; --m) {
    mm_real(A, E, P1, NDIM, NDIM, NDIM, NDIM, fs);
    const double invm = 1.0 / (double)m;
    for (int e = tid; e < NDIM * NDIM; e += NTHREADS)
      E[e] = (((e / NDIM) == (e % NDIM)) ? 1.0 : 0.0) + P1[e] * invm;
  }
  __syncthreads();

  // B^2, B^3, B^5 (B = A/2) and Gh = -(B - B^3/3 + (2/15) B^5)
  mm_real(A, A, P1, NDIM, NDIM, NDIM, NDIM, fs);
  for (int e = tid; e < NDIM * NDIM; e += NTHREADS) P1[e] *= 0.25;   // B^2
  mm_real(P1, A, P2, NDIM, NDIM, NDIM, NDIM, fs);
  for (int e = tid; e < NDIM * NDIM; e += NTHREADS) P2[e] *= 0.5;    // B^3
  mm_real(P2, P1, P3, NDIM, NDIM, NDIM, NDIM, fs);                   // B^5
  for (int e = tid; e < NDIM * NDIM; e += NTHREADS)
    Gh[e] = -(0.5 * A[e] - P2[e] * (1.0 / 3.0) + P3[e] * (2.0 / 15.0));
  __syncthreads();

  // logeta = 0.5 tr(B^2) - (1/12) tr(B^4)
  {
    double t2 = 0.0, t4 = 0.0;
    for (int e = tid; e < NDIM * NDIM; e += NTHREADS) {
      const int r = e / NDIM, c = e % NDIM;
      const double v = P1[e];
      if (r == c) t2 += v;
      t4 += v * P1[c * NDIM + r];
    }
    s_red[tid] = t2; s_red2[tid] = t4; __syncthreads();
    for (int s = NTHREADS / 2; s > 0; s >>= 1) {
      if (tid < s) { s_red[tid] += s_red[tid + s]; s_red2[tid] += s_red2[tid + s]; }
      __syncthreads();
    }
    if (tid == 0) s_le = 0.5 * s_red[0] - (1.0 / 12.0) * s_red2[0];
    __syncthreads();
  }

  // v, R = expH @ v, Gz = green(v, v), Ghz = green(v, R)  (-> shared region)
  build_v(s0, PXv, V);
  mm_rc(E, V, R, NDIM, NDIM, NS, NS, fs);
  green_from_two(V, V, Gz, Lh, Ma, X, fs);
  dc* Ghz = (dc*)(shr + (size_t)px * 8192);
  green_from_two(V, R, Ghz, Lh, Ma, X, fs);

  // logeta_Ghz = le + log(sign_pf) + logabs_pf   (sign_pref = (-1)^32 = +1)
  build_pfmat_r(Gh, Gz, sP);
  slog_pfaffian(sP, resSign, resLa);
  if (tid == 0) {
    const dc lg = clog_(*resSign);
    ((dc*)(shr + 16384))[px] = dc{s_le + lg.re + *resLa, lg.im};
  }
}

// ===========================================================================
// Batch kernel: one block per batch element; two passes (PX=+1 plus-state,
// PX=-1 minus-state), each: green_from_two -> pfmat(LDS) -> Pfaffian; then
// complex logsumexp combine.
// ===========================================================================
__global__ void batch_kernel(const double* __restrict__ x,
                             const double* __restrict__ s0,
                             double* __restrict__ ws,
                             double* __restrict__ out)
{
  __shared__ dc sP[PDIM * PDIM];             // 256 KB LDS Pfaffian matrix
  const int b = blockIdx.x;
  double* shr = ws;
  double* ar  = ws + SHARED_DBL + (size_t)b * ARENA_DBL;
  dc* Gs = (dc*)(ar + GS_OFF);
  dc* ST = (dc*)(ar + ST_OFF);
  dc* V  = (dc*)(ar + VX_OFF);
  dc* Lh = (dc*)(ar + LH_OFF);
  dc* Ma = (dc*)(ar + MA_OFF);
  dc* X  = (dc*)(ar + X_OFF);
  dc* resSign = (dc*)(ar + RES_OFF);
  double* resLa = ar + RES_OFF + 2;
  float* fs = (float*)(ar + FS_OFF);
  const double* xb = x + (size_t)b * NS;
  const int tid = threadIdx.x;
  const double inv_s = 0.70710678118654752440;

  dc lo[2];
  for (int pass = 0; pass < 2; ++pass) {       // 0: plus (PX=+1), 1: minus (PX=-1)
    const int PXv = (pass == 0) ? 1 : -1;
    build_v(xb, PXv, V);
    // state: rows 2k -> phase*norm (plus: -i/sqrt2 all k; minus: +i/sqrt2 at k=n-1),
    //        rows 2k+1 -> 1/sqrt2
    for (int e = tid; e < NDIM * NS; e += NTHREADS) ST[e] = dc{0.0, 0.0};
    __syncthreads();
    for (int k = tid; k < NS; k += NTHREADS) {
      double ph_im = -inv_s;
      if (pass == 1 && k == NS - 1) ph_im = inv_s;
      ST[(2 * k) * NS + k]     = dc{0.0, ph_im};
      ST[(2 * k + 1) * NS + k] = dc{inv_s, 0.0};
    }
    __syncthreads();

    green_from_two(V, ST, Gs, Lh, Ma, X, fs);
    const dc* Ghz = (const dc*)(shr + (size_t)pass * 8192);
    build_pfmat(Ghz, Gs, sP);
    slog_pfaffian(sP, resSign, resLa);
    if (tid == 0) {
      const dc le = ((const dc*)(shr + 16384))[pass];
      const dc lg = clog_(*resSign);
      lo[pass] = dc{le.re + lg.re + *resLa, le.im + lg.im};
    }
    __syncthreads();
  }

  if (tid == 0) {
    const dc lo_p = lo[0], lo_m = lo[1];
    const double prod = xb[NS - 1] * s0[NS - 1];
    const dc lt = clog_(dc{prod, 0.0});        // log(+-1) -> 0 or i*pi
    const dc a = lo_m;
    const dc c = cadd(lo_p, lt);
    const double m = fmax(a.re, c.re);
    const dc s = cadd(cexp_(dc{a.re - m, a.im}), cexp_(dc{c.re - m, c.im}));
    const dc lr = clog_(s);
    ((dc*)out)[b] = dc{m + lr.re, lr.im};
  }
}

// ===========================================================================
extern "C" void kernel_launch(void* const* d_in, const int* in_sizes, int n_in,
                              void* d_out, int out_size, void* d_ws, size_t ws_size,
                              hipStream_t stream)
{
  (void)in_sizes; (void)n_in; (void)out_size; (void)ws_size;
  const double* x  = (const double*)d_in[0];   // (64, 32) f64
  const double* s0 = (const double*)d_in[1];   // (32,)    f64
  const double* H1 = (const double*)d_in[2];   // (64, 64) f64
  const double* H2 = (const double*)d_in[3];   // (64, 64) f64
  double* ws = (double*)d_ws;

  precompute_kernel<<<2, NTHREADS, 0, stream>>>(H1, H2, s0, ws);
  batch_kernel<<<NBATCH, NTHREADS, 0, stream>>>(x, s0, ws, (double*)d_out);
}